// NTMAttention_30983894073572
// MI455X (gfx1250) — compile-verified
//
#include <hip/hip_runtime.h>
#include <math.h>

typedef float v2f __attribute__((ext_vector_type(2)));
typedef float v8f __attribute__((ext_vector_type(8)));

#define B_DIM 256
#define N_DIM 4096
#define M_DIM 64
#define NTHREADS 256
#define NWAVES (NTHREADS / 32)
#define ROWS_PER_THREAD (N_DIM / NTHREADS)   // 16

// ---------------- block-wide reductions (wave32) ----------------
__device__ __forceinline__ float blockSum(float v, float* red) {
#pragma unroll
  for (int off = 16; off > 0; off >>= 1) v += __shfl_xor(v, off, 32);
  const int lane = threadIdx.x & 31, wid = threadIdx.x >> 5;
  if (lane == 0) red[wid] = v;
  __syncthreads();
  float r = (threadIdx.x < NWAVES) ? red[threadIdx.x] : 0.0f;
  if (wid == 0) {
#pragma unroll
    for (int off = NWAVES / 2; off > 0; off >>= 1) r += __shfl_xor(r, off, 32);
    if (lane == 0) red[0] = r;
  }
  __syncthreads();
  float out = red[0];
  __syncthreads();
  return out;
}

__device__ __forceinline__ float blockMax(float v, float* red) {
#pragma unroll
  for (int off = 16; off > 0; off >>= 1) v = fmaxf(v, __shfl_xor(v, off, 32));
  const int lane = threadIdx.x & 31, wid = threadIdx.x >> 5;
  if (lane == 0) red[wid] = v;
  __syncthreads();
  float r = (threadIdx.x < NWAVES) ? red[threadIdx.x] : -INFINITY;
  if (wid == 0) {
#pragma unroll
    for (int off = NWAVES / 2; off > 0; off >>= 1) r = fmaxf(r, __shfl_xor(r, off, 32));
    if (lane == 0) red[0] = r;
  }
  __syncthreads();
  float out = red[0];
  __syncthreads();
  return out;
}

// One workgroup per batch element b. 256 threads = 8 waves.
__global__ __launch_bounds__(NTHREADS)
void ntm_address_kernel(const float* __restrict__ beta,
                        const float* __restrict__ kappa,
                        const float* __restrict__ gam,
                        const float* __restrict__ gvec,
                        const float* __restrict__ svec,
                        const float* __restrict__ w_prev,
                        const float* __restrict__ memory,
                        float* __restrict__ out) {
  __shared__ float s_dot[N_DIM];   // dot -> z -> w_g (reused in place)
  __shared__ float s_na[N_DIM];    // squared row norms
  __shared__ float s_kap[M_DIM];   // kappa + eps
  __shared__ float s_red[NWAVES];  // reduction scratch

  const int b = blockIdx.x;
  const int tid = threadIdx.x;
  const int lane = tid & 31;
  const int wid = tid >> 5;

  if (tid < M_DIM) s_kap[tid] = kappa[b * M_DIM + tid] + 1e-12f;
  __syncthreads();

  // ||kappa + eps||  (redundant per-thread; 64 LDS reads, negligible)
  float nb2 = 0.0f;
#pragma unroll
  for (int m = 0; m < M_DIM; ++m) { const float k = s_kap[m]; nb2 += k * k; }
  const float nbc = fmaxf(sqrtf(nb2), 1e-8f);

  // ---------------- phase 1: stream memory once; WMMA dot + VALU norms ----
  // A fragment (16x4 f32): lane L holds row (L&15); VGPR0 = K(2*half),
  // VGPR1 = K(2*half+1)  -> one float2 load per K-chunk per lane.
  // B fragment broadcast: B[k][j] = kappa_eps[k] for all 16 columns j, so
  // every column of D accumulates the identical row-dot (exact f32 math).
  const float* memb = memory + (size_t)b * N_DIM * M_DIM;
  const int half = lane >> 4;   // 0 or 1
  const int mrow = lane & 15;

  for (int tile = wid; tile < N_DIM / 16; tile += NWAVES) {
    const int n0 = tile * 16;
    const float* rowp = memb + (size_t)(n0 + mrow) * M_DIM + half * 2;
    v8f acc = {0.f, 0.f, 0.f, 0.f, 0.f, 0.f, 0.f, 0.f};
    float sq = 0.0f;
#pragma unroll
    for (int k0 = 0; k0 < M_DIM; k0 += 4) {
      v2f a = *(const v2f*)(rowp + k0);
      a.x += 1e-12f;
      a.y += 1e-12f;
      sq += a.x * a.x + a.y * a.y;
      v2f bf;
      bf.x = s_kap[k0 + 2 * half];
      bf.y = s_kap[k0 + 2 * half + 1];
      acc = __builtin_amdgcn_wmma_f32_16x16x4_f32(false, a, false, bf,
                                                  (short)0, acc, false, false);
    }
    // fold squared norms across lane halves (K 0..1 half | K 2..3 half)
    const float sq2 = sq + __shfl_xor(sq, 16, 32);
    if (lane < 16) s_na[n0 + lane] = sq2;
    // D layout: VGPR r -> row r (lanes 0-15) / row r+8 (lanes 16-31),
    // value replicated across the 16 lanes of each half.
    if (mrow == 0) {
      const int base = n0 + half * 8;
#pragma unroll
      for (int r = 0; r < 8; ++r) s_dot[base + r] = acc[r];
    }
  }
  __syncthreads();

  // ---------------- phase 2: softmax / interpolate / shift / sharpen ------
  const float beta_b = beta[b];
  const float gamma_b = gam[b];
  const float g_b = gvec[b];
  const float s0 = svec[b * 3 + 0];
  const float s1 = svec[b * 3 + 1];
  const float s2 = svec[b * 3 + 2];

  float zloc[ROWS_PER_THREAD];
  float lmax = -INFINITY;
#pragma unroll
  for (int i = 0; i < ROWS_PER_THREAD; ++i) {
    const int n = tid + i * NTHREADS;
    const float dot = s_dot[n];
    const float na = fmaxf(sqrtf(s_na[n]), 1e-8f);
    const float z = beta_b * (dot / (na * nbc));
    zloc[i] = z;
    lmax = fmaxf(lmax, z);
  }
  const float zmax = blockMax(lmax, s_red);

  float lsum = 0.0f;
#pragma unroll
  for (int i = 0; i < ROWS_PER_THREAD; ++i) {
    const float e = __expf(zloc[i] - zmax);
    zloc[i] = e;
    lsum += e;
  }
  const float esum = blockSum(lsum, s_red);
  const float einv = 1.0f / esum;

  // w_g = g*w_c + (1-g)*w_prev  (store into s_dot for the circular shift)
#pragma unroll
  for (int i = 0; i < ROWS_PER_THREAD; ++i) {
    const int n = tid + i * NTHREADS;
    const float wc = zloc[i] * einv;
    s_dot[n] = g_b * wc + (1.0f - g_b) * w_prev[(size_t)b * N_DIM + n];
  }
  __syncthreads();

  // w_hat[i] = s0*w_g[i-1] + s1*w_g[i] + s2*w_g[i+1]  (mod N), then pow(gamma)
  float lsum2 = 0.0f;
#pragma unroll
  for (int i = 0; i < ROWS_PER_THREAD; ++i) {
    const int n = tid + i * NTHREADS;
    const float wh = s0 * s_dot[(n + N_DIM - 1) & (N_DIM - 1)] +
                     s1 * s_dot[n] +
                     s2 * s_dot[(n + 1) & (N_DIM - 1)];
    const float p = powf(wh, gamma_b);
    zloc[i] = p;
    lsum2 += p;
  }
  const float psum = blockSum(lsum2, s_red);
  const float pinv = 1.0f / (psum + 1e-12f);

#pragma unroll
  for (int i = 0; i < ROWS_PER_THREAD; ++i) {
    const int n = tid + i * NTHREADS;
    out[(size_t)b * N_DIM + n] = zloc[i] * pinv;
  }
}

extern "C" void kernel_launch(void* const* d_in, const int* in_sizes, int n_in,
                              void* d_out, int out_size, void* d_ws, size_t ws_size,
                              hipStream_t stream) {
  (void)in_sizes; (void)n_in; (void)out_size; (void)d_ws; (void)ws_size;
  const float* beta   = (const float*)d_in[0];
  const float* kappa  = (const float*)d_in[1];
  const float* gam    = (const float*)d_in[2];
  const float* g      = (const float*)d_in[3];
  const float* s      = (const float*)d_in[4];
  const float* w_prev = (const float*)d_in[5];
  const float* memory = (const float*)d_in[6];
  // d_in[7] = int_shift (==3, compile-time assumed; s has exactly 3 taps)
  float* out = (float*)d_out;

  ntm_address_kernel<<<B_DIM, NTHREADS, 0, stream>>>(
      beta, kappa, gam, g, s, w_prev, memory, out);
}